// LovaszSoftMaxWithCE_27204322853622
// MI455X (gfx1250) — compile-verified
//
#include <hip/hip_runtime.h>
#include <cstdint>

#define NTHR 256
static constexpr int    NC    = 21;
static constexpr int    HW_   = 512 * 512;           // 262144 = 2^18
static constexpr size_t PX    = (size_t)8 * HW_;     // 2097152
static constexpr int    TILE_ = 2048;                // items per sort block
static constexpr int    NB_   = (int)(PX / TILE_);   // 1024 sort blocks
static constexpr int    NBA   = (int)(PX / NTHR);    // 8192 pixel blocks
static constexpr int    IGN   = 255;

// ---------------- workspace layout (bytes) ----------------
static constexpr size_t OFF_ITEMS_A = 0;
static constexpr size_t OFF_ITEMS_B = OFF_ITEMS_A + PX * 8;
static constexpr size_t OFF_LSE     = OFF_ITEMS_B + PX * 8;
static constexpr size_t OFF_HIST    = OFF_LSE + PX * 4;
static constexpr size_t OFF_BINTOT  = OFF_HIST + (size_t)256 * NB_ * 4;
static constexpr size_t OFF_BINBASE = OFF_BINTOT + 256 * 4;
static constexpr size_t OFF_FGBLK   = OFF_BINBASE + 256 * 4;
static constexpr size_t OFF_FGBASE  = OFF_FGBLK + (size_t)NB_ * 4;
static constexpr size_t OFF_CEPART  = OFF_FGBASE + (size_t)NB_ * 4;
static constexpr size_t OFF_VCPART  = OFF_CEPART + (size_t)NBA * 4;
static constexpr size_t OFF_LOSSP   = OFF_VCPART + (size_t)NBA * 4;
static constexpr size_t OFF_SCAL    = OFF_LOSSP + (size_t)NB_ * 4;
// scal[0]=ceSum, scal[1]=validCnt, scal[2+c]=loss_c, scal[23+c]=gts_c

typedef __attribute__((ext_vector_type(8))) int   v8i;
typedef __attribute__((ext_vector_type(8))) float v8f;

#if __has_builtin(__builtin_amdgcn_global_load_async_to_lds_b64)
#define HAVE_ASYNC_LDS 1
typedef int v2i_ __attribute__((vector_size(8)));
typedef __attribute__((address_space(1))) v2i_ as1_v2i;
typedef __attribute__((address_space(3))) v2i_ as3_v2i;
__device__ __forceinline__ void async_g2l_b64(const void* g, const void* l) {
  __builtin_amdgcn_global_load_async_to_lds_b64(
      (as1_v2i*)(uintptr_t)g,
      (as3_v2i*)(unsigned int)(uintptr_t)l, 0, 0);
}
__device__ __forceinline__ void wait_async0() {
#if __has_builtin(__builtin_amdgcn_s_wait_asynccnt)
  __builtin_amdgcn_s_wait_asynccnt(0);
#else
  asm volatile("s_wait_asynccnt 0" ::: "memory");
#endif
}
#endif

// ---------------- helpers ----------------
__device__ __forceinline__ unsigned h2u(_Float16 h) {
  union { _Float16 h; unsigned short u; } c;
  c.h = h;
  return (unsigned)c.u;
}

// Inclusive Hillis–Steele scan of 256 uints in LDS. Caller syncs before call.
__device__ __forceinline__ void scan256_incl(unsigned int* s) {
  int t = threadIdx.x;
#pragma unroll
  for (int o = 1; o < 256; o <<= 1) {
    unsigned v = (t >= o) ? s[t - o] : 0u;
    __syncthreads();
    s[t] += v;
    __syncthreads();
  }
}

// Sum of 256 floats (one per thread). Fixed association -> deterministic.
// WMMA (inline asm): with B = ones, the sum over every element of D equals
// 16 * sum over every element of A, independent of operand lane layout.
// Each f32 value is split into f16 hi + f16 lo (v = hi + lo to ~2^-22 rel),
// so one 16x32 f16 A tile carries all 256 values at near-f32 accuracy
// (accumulation is in the f32 C/D matrix).
__device__ __forceinline__ float block_reduce_sum_256(float v, float* sbuf) {
  int t = threadIdx.x;
  sbuf[t] = v;
  __syncthreads();
  if (t < 32) {  // wave 0 fully active -> EXEC all ones for WMMA
    v8i a, b;
#pragma unroll
    for (int i = 0; i < 8; ++i) b[i] = 0x3C003C00;  // packed f16 {1.0, 1.0}
#pragma unroll
    for (int k = 0; k < 8; ++k) {
      float x = sbuf[t * 8 + k];
      _Float16 hi = (_Float16)x;
      _Float16 lo = (_Float16)(x - (float)hi);
      a[k] = (int)(h2u(hi) | (h2u(lo) << 16));
    }
    v8f c = {0.f, 0.f, 0.f, 0.f, 0.f, 0.f, 0.f, 0.f};
    v8f d;
    // D = A(16x32 f16) * B(32x16 f16, ones) + C ; trailing v_nops cover the
    // WMMA->VALU RAW hazard since the compiler can't see this is a WMMA.
    asm volatile(
        "v_wmma_f32_16x16x32_f16 %0, %1, %2, %3\n\t"
        "v_nop\n\t"
        "v_nop\n\t"
        "v_nop\n\t"
        "v_nop\n\t"
        "v_nop"
        : "=v"(d)
        : "v"(a), "v"(b), "v"(c));
    float s = d[0] + d[1] + d[2] + d[3] + d[4] + d[5] + d[6] + d[7];
    sbuf[t] = s;  // per-lane partial (sum over its 8 D entries)
  }
  __syncthreads();
  if (t == 0) {
    float tot = 0.f;
#pragma unroll
    for (int i = 0; i < 32; ++i) tot += sbuf[i];
    sbuf[0] = tot * 0.0625f;  // sum over all D entries = 16 * total
  }
  __syncthreads();
  float r = sbuf[0];
  __syncthreads();
  return r;
}

// ---------------- kernels ----------------

// Per pixel: logsumexp over 21 classes, CE contribution -> block partials.
__global__ void k_softmax_ce(const float* __restrict__ logits,
                             const int* __restrict__ labels,
                             float* __restrict__ lse,
                             float* __restrict__ cePart,
                             float* __restrict__ vcPart) {
  __shared__ float red[256];
  int t = threadIdx.x;
  int p = blockIdx.x * NTHR + t;
  int b = p >> 18;
  int hw = p & (HW_ - 1);
  const float* base = logits + (size_t)b * NC * HW_ + hw;
  float xv[NC];
#pragma unroll
  for (int c = 0; c < NC; ++c) xv[c] = base[(size_t)c * HW_];
  float m = xv[0];
#pragma unroll
  for (int c = 1; c < NC; ++c) m = fmaxf(m, xv[c]);
  float s = 0.f;
#pragma unroll
  for (int c = 0; c < NC; ++c) s += expf(xv[c] - m);
  float l = m + logf(s);
  lse[p] = l;
  int lab = labels[p];
  bool valid = (lab != IGN);
  int sl = valid ? (lab < 0 ? 0 : (lab > NC - 1 ? NC - 1 : lab)) : 0;
  float xl = base[(size_t)sl * HW_];  // reload: avoids dynamic reg indexing
  float nll = l - xl;
  float ce = block_reduce_sum_256(valid ? nll : 0.f, red);
  float vc = block_reduce_sum_256(valid ? 1.f : 0.f, red);
  if (t == 0) { cePart[blockIdx.x] = ce; vcPart[blockIdx.x] = vc; }
}

// Fixed-order reduction of n block partials (n multiple of 256).
__global__ void k_reduce_part(const float* __restrict__ part, int n,
                              float* __restrict__ out) {
  __shared__ float s[256];
  int t = threadIdx.x;
  int per = n / 256;
  float sum = 0.f;
  for (int k = 0; k < per; ++k) sum += part[t * per + k];
  s[t] = sum;
  __syncthreads();
#pragma unroll
  for (int o = 128; o; o >>= 1) {
    if (t < o) s[t] += s[t + o];
    __syncthreads();
  }
  if (t == 0) out[0] = s[0];
}

// Build 64-bit sort items: key (descending-err transform) << 32 | fg<<1 | valid.
__global__ void k_build_keys(const float* __restrict__ logits,
                             const int* __restrict__ labels,
                             const float* __restrict__ lse,
                             unsigned long long* __restrict__ items, int cls) {
  int p = blockIdx.x * NTHR + threadIdx.x;
  int b = p >> 18;
  int hw = p & (HW_ - 1);
  float x = logits[(size_t)b * NC * HW_ + (size_t)cls * HW_ + hw];
  float pc = expf(x - lse[p]);
  int lab = labels[p];
  unsigned vv = (lab != IGN) ? 1u : 0u;
  unsigned fg = (lab == cls) ? 1u : 0u;
  float err = vv ? fabsf((float)fg - pc) : -1.0f;
  unsigned u = __float_as_uint(err);
  unsigned kasc = (u & 0x80000000u) ? ~u : (u | 0x80000000u);  // ascending xform
  unsigned key = ~kasc;  // ascending radix on key == descending err (stable)
  items[p] = ((unsigned long long)key << 32) | (fg << 1) | vv;
}

// Per-block 256-bin histogram of one radix digit; bin-major output.
__global__ void k_hist(const unsigned long long* __restrict__ src,
                       unsigned int* __restrict__ hist, int shift) {
  __shared__ unsigned int lh[256];
#if defined(HAVE_ASYNC_LDS)
  __shared__ unsigned long long tile[TILE_];
#endif
  int t = threadIdx.x, blk = blockIdx.x;
  lh[t] = 0;
  size_t base = (size_t)blk * TILE_;
#if defined(HAVE_ASYNC_LDS)
#pragma unroll
  for (int j = 0; j < 8; ++j) {
    size_t idx = base + (size_t)j * NTHR + t;
    async_g2l_b64((const void*)(src + idx), (const void*)(tile + j * NTHR + t));
  }
  wait_async0();
  __syncthreads();
#pragma unroll
  for (int j = 0; j < 8; ++j) {
    unsigned d = (unsigned)(tile[j * NTHR + t] >> shift) & 255u;
    atomicAdd(&lh[d], 1u);
  }
#else
  __syncthreads();
#pragma unroll
  for (int j = 0; j < 8; ++j) {
    unsigned long long v = src[base + (size_t)j * NTHR + t];
    unsigned d = (unsigned)(v >> shift) & 255u;
    atomicAdd(&lh[d], 1u);
  }
#endif
  __syncthreads();
  hist[(size_t)t * NB_ + blk] = lh[t];
}

// Per-bin totals across the NB source blocks.
__global__ void k_bin_total(const unsigned int* __restrict__ hist,
                            unsigned int* __restrict__ binTot) {
  __shared__ unsigned int s[256];
  int t = threadIdx.x, b = blockIdx.x;
  unsigned sum = 0;
#pragma unroll
  for (int k = 0; k < NB_ / NTHR; ++k) sum += hist[(size_t)b * NB_ + k * NTHR + t];
  s[t] = sum;
  __syncthreads();
#pragma unroll
  for (int o = 128; o; o >>= 1) {
    if (t < o) s[t] += s[t + o];
    __syncthreads();
  }
  if (t == 0) binTot[b] = s[0];
}

// Exclusive scan of the 256 bin totals.
__global__ void k_bin_base(const unsigned int* __restrict__ binTot,
                           unsigned int* __restrict__ binBase) {
  __shared__ unsigned int s[256];
  int t = threadIdx.x;
  unsigned orig = binTot[t];
  s[t] = orig;
  __syncthreads();
  scan256_incl(s);
  binBase[t] = s[t] - orig;
}

// In-place: hist[bin*NB + blk] -> global exclusive offset for (bin, blk).
__global__ void k_bin_offsets(unsigned int* __restrict__ hist,
                              const unsigned int* __restrict__ binBase) {
  __shared__ unsigned int s[256];
  int t = threadIdx.x, b = blockIdx.x;
  size_t base = (size_t)b * NB_ + (size_t)t * 4;
  unsigned loc[4]; unsigned sum = 0;
#pragma unroll
  for (int k = 0; k < 4; ++k) { loc[k] = hist[base + k]; sum += loc[k]; }
  s[t] = sum;
  __syncthreads();
  scan256_incl(s);
  unsigned run = binBase[b] + s[t] - sum;
#pragma unroll
  for (int k = 0; k < 4; ++k) { hist[base + k] = run; run += loc[k]; }
}

// Stable scatter: rounds in memory order, waves serialized, ballot ranking.
__global__ void k_scatter(const unsigned long long* __restrict__ src,
                          unsigned long long* __restrict__ dst,
                          const unsigned int* __restrict__ offs, int shift) {
  __shared__ unsigned int off[256];
  __shared__ unsigned int cnt[256];
  int t = threadIdx.x, blk = blockIdx.x;
  off[t] = offs[(size_t)t * NB_ + blk];
  cnt[t] = 0;
  __syncthreads();
  int lane = t & 31, wave = t >> 5;
  unsigned lt = (1u << lane) - 1u;
  size_t base = (size_t)blk * TILE_;
  for (int j = 0; j < 8; ++j) {
    unsigned long long v = src[base + (size_t)j * NTHR + t];
    if (j < 7) __builtin_prefetch((const void*)&src[base + (size_t)(j + 1) * NTHR + t], 0, 1);
    unsigned d = (unsigned)(v >> shift) & 255u;
    for (int w = 0; w < 8; ++w) {
      if (wave == w) {
        unsigned mask = 0xFFFFFFFFu;
#pragma unroll
        for (int bit = 0; bit < 8; ++bit) {
          unsigned bal = (unsigned)__ballot((int)((d >> bit) & 1u));
          mask &= ((d >> bit) & 1u) ? bal : ~bal;
        }
        unsigned rank = __popc(mask & lt);
        unsigned prev = cnt[d];
        dst[off[d] + prev + rank] = v;
        if ((mask & lt) == 0u) cnt[d] = prev + __popc(mask);  // group leader
      }
      __syncthreads();
    }
  }
}

// Per-block fg-bit sums over sorted items.
__global__ void k_fg_block(const unsigned long long* __restrict__ items,
                           unsigned int* __restrict__ fgBlk) {
  __shared__ unsigned int s[256];
  int t = threadIdx.x, blk = blockIdx.x;
  size_t base = (size_t)blk * TILE_;
  unsigned sum = 0;
#pragma unroll
  for (int j = 0; j < 8; ++j) sum += (unsigned)(items[base + (size_t)j * NTHR + t] >> 1) & 1u;
  s[t] = sum;
  __syncthreads();
#pragma unroll
  for (int o = 128; o; o >>= 1) {
    if (t < o) s[t] += s[t + o];
    __syncthreads();
  }
  if (t == 0) fgBlk[blk] = s[0];
}

// Exclusive scan of 1024 block sums; also records gts (= total fg) per class.
__global__ void k_fg_scan(const unsigned int* __restrict__ fgBlk,
                          unsigned int* __restrict__ fgBase,
                          float* __restrict__ scal, int cls) {
  __shared__ unsigned int s[256];
  int t = threadIdx.x;
  unsigned loc[4]; unsigned sum = 0;
#pragma unroll
  for (int k = 0; k < 4; ++k) { loc[k] = fgBlk[t * 4 + k]; sum += loc[k]; }
  s[t] = sum;
  __syncthreads();
  scan256_incl(s);
  unsigned run = s[t] - sum;
#pragma unroll
  for (int k = 0; k < 4; ++k) { fgBase[t * 4 + k] = run; run += loc[k]; }
  if (t == 255) scal[23 + cls] = (float)s[255];
}

// Lovasz gradient dot product over sorted items; local grad from prefix count.
// Deliberately de-exoticized for this toolchain's ISel: rolled loops (no wide
// load vectorization, no SLP packing), laundered per-lane VMEM loads instead
// of uniform SMEM loads, rcp-mul instead of f32 division, plain LDS tree
// reduction (the WMMA reduction lives in k_softmax_ce).
__global__ void k_lovasz_reduce(const unsigned long long* __restrict__ items,
                                const unsigned int* __restrict__ fgBase,
                                const float* __restrict__ scal, int cls,
                                float* __restrict__ lossPart) {
  __shared__ unsigned int ts[256];
  __shared__ float red[256];
  int t = threadIdx.x, blk = blockIdx.x;

  // Launder uniform indices into VGPRs -> plain per-lane global_load_b32.
  int gidx = 23 + cls;
  asm volatile("" : "+v"(gidx));
  float gts = scal[gidx];
  int bidx = blk;
  asm volatile("" : "+v"(bidx));
  unsigned cfbase = fgBase[bidx];

  unsigned base32 = (unsigned)blk * (unsigned)TILE_ + (unsigned)t * 8u;
  const unsigned long long* ip = items + base32;  // consecutive per thread
  unsigned fsum = 0;
#pragma unroll 1
  for (int k = 0; k < 8; ++k) fsum += (unsigned)(ip[k] >> 1) & 1u;
  ts[t] = fsum;
  __syncthreads();
  scan256_incl(ts);
  unsigned cf = cfbase + ts[t] - fsum;  // exclusive prefix at thread start
  float acc = 0.f;
#pragma unroll 1
  for (int k = 0; k < 8; ++k) {
    unsigned long long v = ip[k];  // re-read (L0-resident)
    unsigned fg = (unsigned)(v >> 1) & 1u;
    unsigned vv = (unsigned)v & 1u;
    cf += fg;  // inclusive prefix at this sorted position
    unsigned gi = base32 + (unsigned)k;  // global sorted index (32-bit)
    unsigned key = (unsigned)(v >> 32);
    unsigned kasc = ~key;
    unsigned u = (kasc & 0x80000000u) ? (kasc & 0x7FFFFFFFu) : ~kasc;
    float err = __uint_as_float(u);
    float cff = (float)cf;
    float jac = 1.f - (gts - cff) *
                __builtin_amdgcn_rcpf(gts + (float)(gi + 1u) - cff);
    float grad;
    if (gi == 0u) {
      grad = jac;
    } else {
      float cfm = cff - (float)fg;
      float jacm = 1.f - (gts - cfm) *
                   __builtin_amdgcn_rcpf(gts + (float)gi - cfm);
      grad = jac - jacm;
    }
    acc += (vv ? err : 0.f) * grad;
  }
  red[t] = acc;
  __syncthreads();
#pragma unroll
  for (int o = 128; o; o >>= 1) {
    if (t < o) red[t] += red[t + o];
    __syncthreads();
  }
  if (t == 0) lossPart[blk] = red[0];
}

__global__ void k_finalize(const float* __restrict__ scal, float* __restrict__ out) {
  if (threadIdx.x == 0 && blockIdx.x == 0) {
    float ps = 0.f, ls = 0.f;
    for (int c = 0; c < NC; ++c) {
      float pres = (scal[23 + c] > 0.f) ? 1.f : 0.f;
      ls += scal[2 + c] * pres;
      ps += pres;
    }
    float lov = ls * __builtin_amdgcn_rcpf(fmaxf(ps, 1.f));
    float ce = scal[0] * __builtin_amdgcn_rcpf(fmaxf(scal[1], 1.f));
    out[0] = lov + ce;  // ALPHA = 1
  }
}

// ---------------- launch ----------------
extern "C" void kernel_launch(void* const* d_in, const int* in_sizes, int n_in,
                              void* d_out, int out_size, void* d_ws, size_t ws_size,
                              hipStream_t stream) {
  (void)in_sizes; (void)n_in; (void)out_size; (void)ws_size;
  const float* logits = (const float*)d_in[0];
  const int* labels = (const int*)d_in[1];
  char* ws = (char*)d_ws;
  unsigned long long* itemsA = (unsigned long long*)(ws + OFF_ITEMS_A);
  unsigned long long* itemsB = (unsigned long long*)(ws + OFF_ITEMS_B);
  float* lse = (float*)(ws + OFF_LSE);
  unsigned int* hist = (unsigned int*)(ws + OFF_HIST);
  unsigned int* binTot = (unsigned int*)(ws + OFF_BINTOT);
  unsigned int* binBase = (unsigned int*)(ws + OFF_BINBASE);
  unsigned int* fgBlk = (unsigned int*)(ws + OFF_FGBLK);
  unsigned int* fgBase = (unsigned int*)(ws + OFF_FGBASE);
  float* cePart = (float*)(ws + OFF_CEPART);
  float* vcPart = (float*)(ws + OFF_VCPART);
  float* lossPart = (float*)(ws + OFF_LOSSP);
  float* scal = (float*)(ws + OFF_SCAL);

  k_softmax_ce<<<NBA, NTHR, 0, stream>>>(logits, labels, lse, cePart, vcPart);
  k_reduce_part<<<1, NTHR, 0, stream>>>(cePart, NBA, &scal[0]);
  k_reduce_part<<<1, NTHR, 0, stream>>>(vcPart, NBA, &scal[1]);

  for (int c = 0; c < NC; ++c) {
    k_build_keys<<<NBA, NTHR, 0, stream>>>(logits, labels, lse, itemsA, c);
    unsigned long long* src = itemsA;
    unsigned long long* dst = itemsB;
    for (int pass = 0; pass < 4; ++pass) {
      int shift = 32 + 8 * pass;
      k_hist<<<NB_, NTHR, 0, stream>>>(src, hist, shift);
      k_bin_total<<<256, NTHR, 0, stream>>>(hist, binTot);
      k_bin_base<<<1, NTHR, 0, stream>>>(binTot, binBase);
      k_bin_offsets<<<256, NTHR, 0, stream>>>(hist, binBase);
      k_scatter<<<NB_, NTHR, 0, stream>>>(src, dst, hist, shift);
      unsigned long long* tmp = src; src = dst; dst = tmp;
    }
    // 4 passes -> sorted data back in itemsA
    k_fg_block<<<NB_, NTHR, 0, stream>>>(itemsA, fgBlk);
    k_fg_scan<<<1, NTHR, 0, stream>>>(fgBlk, fgBase, scal, c);
    k_lovasz_reduce<<<NB_, NTHR, 0, stream>>>(itemsA, fgBase, scal, c, lossPart);
    k_reduce_part<<<1, NTHR, 0, stream>>>(lossPart, NB_, &scal[2 + c]);
  }
  k_finalize<<<1, 32, 0, stream>>>(scal, (float*)d_out);
}